// Rules_44358422233254
// MI455X (gfx1250) — compile-verified
//
#include <hip/hip_runtime.h>
#include <hip/hip_bf16.h>

// out[b,u,r] = prod_f( x[b,u,f] * rules[r,f] )
//            = (prod_f x[b,u,f]) * (prod_f rules[r,f])  -- exact factorization
// => rank-1 outer product px[BU] (x) pr[R], done on the WMMA units.
// Roofline: 4MB output store dominates (23.3 TB/s -> ~0.22us floor); all
// multiplies ride on V_WMMA_F32_16X16X4_F32, one per 16x16 output tile.

#define BU   4096   // B*U = 8*512
#define RNUM 256    // rules
#define FNUM 64     // fuzzy sets per rule

typedef float v2f __attribute__((ext_vector_type(2)));
typedef float v8f __attribute__((ext_vector_type(8)));

// --- Stage 1: product-reduce 64 contiguous floats per row --------------------
__global__ void Rules_prod_rows(const float* __restrict__ src,
                                float* __restrict__ dst, int nrows) {
    int row = blockIdx.x * blockDim.x + threadIdx.x;
    if (row >= nrows) return;
    const float4* p = (const float4*)(src + (size_t)row * FNUM);
    float prod = 1.0f;
#pragma unroll
    for (int i = 0; i < FNUM / 4; ++i) {
        float4 v = p[i];
        prod *= v.x * v.y * v.z * v.w;
    }
    dst[row] = prod;
}

// --- Stage 2: outer product px (x) pr via V_WMMA_F32_16X16X4_F32 -------------
//   A (16x4 f32, 2 VGPRs): lane L holds M = L%16 for its K slots.
//     px[M] replicated into ALL K slots -> A[M,k] = px[M] for every k.
//   B (4x16 f32, 2 VGPRs): VGPR0 lanes 0-15 = (K=0, N=lane).
//     pr[N] at K=0, zero everywhere else (value-masked, no exec branch).
//   => D[M,N] = px[M]*pr[N] + px[M]*0 + ...  (exact: zero addends).
//   D (16x16 f32, 8 VGPRs): VGPR v, lane l -> M = v + 8*(l/16), N = l%16.
__global__ void __launch_bounds__(128)
Rules_outer_wmma(const float* __restrict__ px, const float* __restrict__ pr,
                 float* __restrict__ out) {
    const int lane = threadIdx.x & 31;
    const int wave = threadIdx.x >> 5;          // 0..3
    const int m    = lane & 15;
    const bool lo  = (lane < 16);
    const int buBase = blockIdx.x * 16;         // 256 row tiles

    const float a = px[buBase + m];
    v2f A; A[0] = a; A[1] = a;                  // px in every K slot of this lane

    // Unconditional, clause-friendly loads; mask the VALUE (v_cndmask),
    // so EXEC stays all-1s throughout (WMMA requirement, no branches).
    float bq[4];
#pragma unroll
    for (int t = 0; t < 4; ++t)
        bq[t] = pr[(wave * 4 + t) * 16 + m];

#pragma unroll
    for (int t = 0; t < 4; ++t) {
        const int rBase = (wave * 4 + t) * 16;  // 16 col tiles / 4 waves
        v2f Bv; Bv[0] = lo ? bq[t] : 0.0f; Bv[1] = 0.0f;
        v8f c = {};
        // (neg_a, A, neg_b, B, c_mod, C, reuse_a, reuse_b)
        v8f d = __builtin_amdgcn_wmma_f32_16x16x4_f32(
            false, A, false, Bv, (short)0, c, false, false);

        float* o = out + (size_t)(buBase + ((lane >> 4) << 3)) * RNUM + rBase + m;
#pragma unroll
        for (int v = 0; v < 8; ++v)
            o[(size_t)v * RNUM] = d[v];         // two contiguous 64B rows / store
    }
}

extern "C" void kernel_launch(void* const* d_in, const int* in_sizes, int n_in,
                              void* d_out, int out_size, void* d_ws, size_t ws_size,
                              hipStream_t stream) {
    const float* x     = (const float*)d_in[0];   // [B,U,F] f32
    const float* rules = (const float*)d_in[1];   // [R,F]  f32
    // d_in[2] = epoch scalar: unused (eval mode, rules fixed)
    float* out = (float*)d_out;                   // [B,U,R] f32

    float* px = (float*)d_ws;                     // BU floats
    float* pr = px + BU;                          // RNUM floats  (17.4 KB total)

    Rules_prod_rows<<<(BU + 255) / 256, 256, 0, stream>>>(x, px, BU);
    Rules_prod_rows<<<1, 256, 0, stream>>>(rules, pr, RNUM);
    Rules_outer_wmma<<<BU / 16, 128, 0, stream>>>(px, pr, out);
}